// SigTKANCell_53815940219116
// MI455X (gfx1250) — compile-verified
//
#include <hip/hip_runtime.h>
#include <math.h>

#define B_ 64
#define S_ 512
#define F_ 128
#define U_ 512
#define SIGDIM 16512          // F + F*F
#define NB 8
#define T_ 511                // S-1
#define KAUG 1152             // F + F*NB
#define KSPLIT 16
#define KCHUNK (SIGDIM / KSPLIT)   // 1032, multiple of 4

typedef float v2f __attribute__((ext_vector_type(2)));
typedef float v8f __attribute__((ext_vector_type(8)));

__device__ __forceinline__ v8f wmma4(v2f a, v2f b, v8f c) {
  // V_WMMA_F32_16X16X4_F32 : D = A(16x4) x B(4x16) + C(16x16), fp32 throughout
  return __builtin_amdgcn_wmma_f32_16x16x4_f32(false, a, false, b, (short)0, c,
                                               false, false);
}

__device__ __forceinline__ float sigm(float x) { return 1.0f / (1.0f + expf(-x)); }
__device__ __forceinline__ float act_apply(float x, int act) {
  if (act == 1) return (x > 0.f) ? x : (expf(x) - 1.f);  // elu
  if (act == 2) return sigm(x);                           // sigmoid
  if (act == 3) return tanhf(x);                          // tanh
  return x;
}

// ---------------------------------------------------------------- weighted
__global__ void k_weighted(const float* __restrict__ inp,
                           const float* __restrict__ tk,
                           float* __restrict__ w, int n) {
  int i = blockIdx.x * blockDim.x + threadIdx.x;
  if (i >= n) return;
  int s = (i / F_) % S_;
  w[i] = inp[i] * tk[s];
}

// ------------------------------------------- signature prep: P, D, s1(->sig)
__global__ void k_pd(const float* __restrict__ w, float* __restrict__ P,
                     float* __restrict__ D, float* __restrict__ sig) {
  int i = blockIdx.x * blockDim.x + threadIdx.x;   // (b,f)
  if (i >= B_ * F_) return;
  int b = i / F_, f = i % F_;
  const float* wb = w + (size_t)b * S_ * F_ + f;
  float acc = 0.f;
  float prev = wb[0];
  for (int t = 0; t < T_; ++t) {
    float cur = wb[(size_t)(t + 1) * F_];
    float dx = cur - prev;
    P[((size_t)b * T_ + t) * F_ + f] = acc + 0.5f * dx;  // prev_cumsum + dx/2
    D[((size_t)b * T_ + t) * F_ + f] = dx;
    acc += dx;
    prev = cur;
  }
  sig[(size_t)b * SIGDIM + f] = acc;                     // s1 (telescoped)
}

// ----------------------------------- s2[b] = P[b]^T @ D[b]  (128x128, K=511)
__global__ void k_sig2(const float* __restrict__ P, const float* __restrict__ D,
                       float* __restrict__ sig) {
  int lane = threadIdx.x;
  int j0 = blockIdx.x * 16;
  int i0 = blockIdx.y * 16;
  int b = blockIdx.z;
  int half = lane >> 4;
  int l15 = lane & 15;
  int kb = half * 2;                 // lanes 0-15: K 0,1 ; lanes 16-31: K 2,3
  const float* pa = P + (size_t)b * T_ * F_ + (size_t)kb * F_ + i0 + l15;
  const float* pb = D + (size_t)b * T_ * F_ + (size_t)kb * F_ + j0 + l15;
  v8f c = {};
  // 127 guard-free steps (K = 508)
  for (int k0 = 0; k0 < T_ - 3; k0 += 4) {
    v2f a, bb;
    a.x = pa[(size_t)k0 * F_];
    a.y = pa[(size_t)k0 * F_ + F_];
    bb.x = pb[(size_t)k0 * F_];
    bb.y = pb[(size_t)k0 * F_ + F_];
    c = wmma4(a, bb, c);
  }
  // remainder step k0 = 508 (covers K = 508..510; 511 padded with zero)
  {
    v2f a, bb;
    a.x = pa[(size_t)508 * F_];
    a.y = (kb == 0) ? pa[(size_t)509 * F_] : 0.f;
    bb.x = pb[(size_t)508 * F_];
    bb.y = (kb == 0) ? pb[(size_t)509 * F_] : 0.f;
    c = wmma4(a, bb, c);
  }
  float* out = sig + (size_t)b * SIGDIM + F_;
#pragma unroll
  for (int v = 0; v < 8; ++v) {
    int row = i0 + v + 8 * half;
    out[(size_t)row * F_ + j0 + l15] = c[v];
  }
}

// ------------- generic C = act(A(MxK) @ B(KxN) + bias); K % 4 == 0, 1 wave/tile
__global__ void k_gemm(const float* __restrict__ A, int lda,
                       const float* __restrict__ Bm, int ldb,
                       const float* __restrict__ bias, float* __restrict__ C,
                       int ldc, int K, int act) {
  int lane = threadIdx.x;
  int n0 = blockIdx.x * 16;
  int m0 = blockIdx.y * 16;
  int half = lane >> 4;
  int l15 = lane & 15;
  int kb = half * 2;
  const float* arow = A + (size_t)(m0 + l15) * lda + kb;     // 8B aligned (lda even)
  const float* bcol = Bm + (size_t)kb * ldb + n0 + l15;
  v8f c = {};
  for (int k0 = 0; k0 < K; k0 += 4) {
    v2f a = *(const v2f*)(arow + k0);                        // global_load_b64
    v2f b;
    b.x = bcol[(size_t)k0 * ldb];
    b.y = bcol[(size_t)k0 * ldb + ldb];
    c = wmma4(a, b, c);
  }
#pragma unroll
  for (int v = 0; v < 8; ++v) {
    int row = m0 + v + 8 * half;
    int col = n0 + l15;
    C[(size_t)row * ldc + col] = act_apply(c[v] + bias[col], act);
  }
}

// ------------- K-split GEMM for the SIGDIM matmuls: partials[ks][64][512]
__global__ void k_gemm_split(const float* __restrict__ A, int lda,
                             const float* __restrict__ Bm, int ldb,
                             float* __restrict__ part) {
  int lane = threadIdx.x;
  int n0 = blockIdx.x * 16;
  int m0 = blockIdx.y * 16;
  int ks = blockIdx.z;
  int half = lane >> 4;
  int l15 = lane & 15;
  int kb = half * 2;
  int kstart = ks * KCHUNK;
  const float* arow = A + (size_t)(m0 + l15) * lda + kstart + kb;
  const float* bcol = Bm + (size_t)(kstart + kb) * ldb + n0 + l15;
  v8f c = {};
  for (int k0 = 0; k0 < KCHUNK; k0 += 4) {
    v2f a = *(const v2f*)(arow + k0);
    v2f b;
    b.x = bcol[(size_t)k0 * ldb];
    b.y = bcol[(size_t)k0 * ldb + ldb];
    c = wmma4(a, b, c);
  }
#pragma unroll
  for (int v = 0; v < 8; ++v) {
    int row = m0 + v + 8 * half;
    int col = n0 + l15;
    part[((size_t)ks * B_ + row) * U_ + col] = c[v];
  }
}

__global__ void k_combine(const float* __restrict__ part,
                          const float* __restrict__ bias,
                          float* __restrict__ C, int act) {
  int i = blockIdx.x * blockDim.x + threadIdx.x;
  if (i >= B_ * U_) return;
  int u = i % U_;
  float acc = 0.f;
#pragma unroll
  for (int ks = 0; ks < KSPLIT; ++ks) acc += part[(size_t)ks * B_ * U_ + i];
  C[i] = act_apply(acc + bias[u], act);
}

// ------------------------- GRKAN tail: y = ysig + t3*t4 (t3 pre-sigmoided),
//                           LayerNorm(eps=1e-3), softmax -> attn
__global__ void k_grkan_fin(const float* __restrict__ ysig,
                            const float* __restrict__ t3,
                            const float* __restrict__ t4,
                            const float* __restrict__ gamma,
                            const float* __restrict__ beta,
                            float* __restrict__ attn) {
  int b = blockIdx.x;
  int tid = threadIdx.x;  // 256 threads, 2 elems each
  __shared__ float red[256];
  int u0 = tid, u1 = tid + 256;
  size_t base = (size_t)b * U_;
  float y0 = ysig[base + u0] + t3[base + u0] * t4[base + u0];
  float y1 = ysig[base + u1] + t3[base + u1] * t4[base + u1];

  red[tid] = y0 + y1;
  __syncthreads();
  for (int s = 128; s > 0; s >>= 1) { if (tid < s) red[tid] += red[tid + s]; __syncthreads(); }
  float mu = red[0] * (1.0f / U_);
  __syncthreads();

  red[tid] = (y0 - mu) * (y0 - mu) + (y1 - mu) * (y1 - mu);
  __syncthreads();
  for (int s = 128; s > 0; s >>= 1) { if (tid < s) red[tid] += red[tid + s]; __syncthreads(); }
  float var = red[0] * (1.0f / U_);
  __syncthreads();

  float inv = rsqrtf(var + 1e-3f);
  y0 = (y0 - mu) * inv * gamma[u0] + beta[u0];
  y1 = (y1 - mu) * inv * gamma[u1] + beta[u1];

  red[tid] = fmaxf(y0, y1);
  __syncthreads();
  for (int s = 128; s > 0; s >>= 1) { if (tid < s) red[tid] = fmaxf(red[tid], red[tid + s]); __syncthreads(); }
  float mx = red[0];
  __syncthreads();

  float e0 = expf(y0 - mx), e1 = expf(y1 - mx);
  red[tid] = e0 + e1;
  __syncthreads();
  for (int s = 128; s > 0; s >>= 1) { if (tid < s) red[tid] += red[tid + s]; __syncthreads(); }
  float dn = 1.0f / red[0];
  attn[base + u0] = e0 * dn;
  attn[base + u1] = e1 * dn;
}

// ----------------- B_aug = [base_w (128xU) ; spline_w^T ((F*NB)xU)]
__global__ void k_baug(const float* __restrict__ base_w,
                       const float* __restrict__ spline_w,
                       float* __restrict__ Baug) {
  int i = blockIdx.x * blockDim.x + threadIdx.x;
  if (i >= KAUG * U_) return;
  int r = i / U_, u = i % U_;
  float v;
  if (r < F_) {
    v = base_w[(size_t)r * U_ + u];
  } else {
    int fk = r - F_;
    int f = fk >> 3, k = fk & 7;                 // spline_w: (U,F,NB)
    v = spline_w[((size_t)u * F_ + f) * NB + k];
  }
  Baug[i] = v;
}

// -------- fused KAN: build A_aug tile in LDS (silu + b-splines), WMMA GEMM
//          vs B_aug, then attn-weighted column sums -> partials (deterministic)
__global__ void __launch_bounds__(256) k_kan(const float* __restrict__ w,
                                             const float* __restrict__ Baug,
                                             const float* __restrict__ attn,
                                             float* __restrict__ partials) {
  extern __shared__ float smem[];
  float* At = smem;                 // 16 x KAUG
  float* scale = smem + 16 * KAUG;  // 16
  int rt = blockIdx.x;              // 16-row tile of N = B*S rows
  int r0 = rt * 16;
  int b = r0 >> 9;                  // S_ == 512 rows per batch
  int s0 = r0 & (S_ - 1);
  int tid = threadIdx.x;

  // phase 1: A_aug tile
  for (int p = tid; p < 16 * F_; p += 256) {
    int m = p >> 7, f = p & 127;
    float x = w[(size_t)(r0 + m) * F_ + f];
    At[m * KAUG + f] = x * sigm(x);  // silu branch
    // b-splines: grid g[i] = -2.2 + 0.4*i (i=0..11), Cox-de Boor to order 3
    float bb[11];
#pragma unroll
    for (int j = 0; j < 11; ++j) {
      float g0 = -2.2f + 0.4f * j;
      bb[j] = (x >= g0 && x < g0 + 0.4f) ? 1.0f : 0.0f;
    }
#pragma unroll
    for (int q = 1; q <= 3; ++q) {
      float inv = 1.0f / (0.4f * q);
      for (int j = 0; j < 11 - q; ++j) {
        float gj = -2.2f + 0.4f * j;
        float gr = gj + 0.4f * (q + 1);
        bb[j] = (x - gj) * inv * bb[j] + (gr - x) * inv * bb[j + 1];
      }
    }
    float* dst = At + m * KAUG + F_ + f * NB;
#pragma unroll
    for (int k = 0; k < NB; ++k) dst[k] = bb[k];
  }
  if (tid < 16) scale[tid] = attn[(size_t)b * U_ + s0 + tid] * (1.0f / S_);
  __syncthreads();

  // phase 2: 8 waves x 4 n-tiles = full N=512
  int wave = tid >> 5, lane = tid & 31;
  int half = lane >> 4, l15 = lane & 15, kb = half * 2;
  const float* arow = At + l15 * KAUG + kb;                 // ds_load_b64
  for (int it = 0; it < 4; ++it) {
    int n0 = (wave * 4 + it) * 16;
    const float* bcol = Baug + (size_t)kb * U_ + n0 + l15;
    v8f c = {};
    for (int k0 = 0; k0 < KAUG; k0 += 4) {     // KAUG % 4 == 0, guard-free
      v2f a = *(const v2f*)(arow + k0);
      v2f bv;
      bv.x = bcol[(size_t)k0 * U_];
      bv.y = bcol[(size_t)k0 * U_ + U_];
      c = wmma4(a, bv, c);
    }
    // phase 3: attn-weighted mean contribution of this 16-row tile
    float partial = 0.f;
#pragma unroll
    for (int v = 0; v < 8; ++v) partial += scale[v + 8 * half] * c[v];
    partial += __shfl_xor(partial, 16, 32);    // combine M-halves per column
    if (half == 0) partials[(size_t)rt * U_ + n0 + l15] = partial;
  }
}

// ------------- current[b,u] = sum over the 32 row-tiles of batch b
__global__ void k_reduce(const float* __restrict__ partials,
                         float* __restrict__ current) {
  int i = blockIdx.x * blockDim.x + threadIdx.x;
  if (i >= B_ * U_) return;
  int b = i / U_, u = i % U_;
  float acc = 0.f;
#pragma unroll
  for (int t = 0; t < 32; ++t) acc += partials[((size_t)(b * 32 + t)) * U_ + u];
  current[i] = acc;
}

// ------------- combined = [current, h_prev]
__global__ void k_combined(const float* __restrict__ current,
                           const float* __restrict__ h_prev,
                           float* __restrict__ comb) {
  int i = blockIdx.x * blockDim.x + threadIdx.x;
  if (i >= B_ * 2 * U_) return;
  int b = i >> 10, u = i & 1023;
  comb[i] = (u < U_) ? current[(size_t)b * U_ + u]
                     : h_prev[(size_t)b * U_ + (u - U_)];
}

// ------------- all 4 LSTM gate GEMMs in one launch (grid.z = gate)
__global__ void k_gates(const float* __restrict__ comb,
                        const float* __restrict__ wf, const float* __restrict__ wi,
                        const float* __restrict__ wc, const float* __restrict__ wo,
                        const float* __restrict__ bf, const float* __restrict__ bi,
                        const float* __restrict__ bc, const float* __restrict__ bo,
                        float* __restrict__ gates) {
  int g = blockIdx.z;
  const float* W = (g == 0) ? wf : (g == 1) ? wi : (g == 2) ? wc : wo;
  const float* bias = (g == 0) ? bf : (g == 1) ? bi : (g == 2) ? bc : bo;
  int act = (g == 2) ? 3 : 2;        // cand: tanh, others: sigmoid
  int lane = threadIdx.x;
  int n0 = blockIdx.x * 16;
  int m0 = blockIdx.y * 16;
  int half = lane >> 4;
  int l15 = lane & 15;
  int kb = half * 2;
  const float* arow = comb + (size_t)(m0 + l15) * (2 * U_) + kb;
  const float* bcol = W + (size_t)kb * U_ + n0 + l15;
  v8f c = {};
  for (int k0 = 0; k0 < 2 * U_; k0 += 4) {
    v2f a = *(const v2f*)(arow + k0);
    v2f b;
    b.x = bcol[(size_t)k0 * U_];
    b.y = bcol[(size_t)k0 * U_ + U_];
    c = wmma4(a, b, c);
  }
  float* out = gates + (size_t)g * B_ * U_;
#pragma unroll
  for (int v = 0; v < 8; ++v) {
    int row = m0 + v + 8 * half;
    int col = n0 + l15;
    out[(size_t)row * U_ + col] = act_apply(c[v] + bias[col], act);
  }
}

// ------------- LSTM tail (gates already activation-fused)
__global__ void k_lstm(const float* __restrict__ gates,
                       const float* __restrict__ c_prev, float* __restrict__ out) {
  int i = blockIdx.x * blockDim.x + threadIdx.x;
  if (i >= B_ * U_) return;
  const float* fg = gates;
  const float* ig = gates + (size_t)B_ * U_;
  const float* cg = gates + (size_t)2 * B_ * U_;
  const float* og = gates + (size_t)3 * B_ * U_;
  float c_new = fg[i] * c_prev[i] + ig[i] * cg[i];
  out[i] = og[i] * tanhf(c_new);   // h_new
  out[B_ * U_ + i] = c_new;        // c_new
}

extern "C" void kernel_launch(void* const* d_in, const int* in_sizes, int n_in,
                              void* d_out, int out_size, void* d_ws, size_t ws_size,
                              hipStream_t stream) {
  (void)in_sizes; (void)n_in; (void)out_size; (void)ws_size;
  const float* inp      = (const float*)d_in[0];
  const float* h_prev   = (const float*)d_in[1];
  const float* c_prev   = (const float*)d_in[2];
  const float* tk       = (const float*)d_in[3];
  const float* base_w   = (const float*)d_in[4];
  const float* spline_w = (const float*)d_in[5];
  const float* g_w1     = (const float*)d_in[6];
  const float* g_b1     = (const float*)d_in[7];
  const float* g_w2     = (const float*)d_in[8];
  const float* g_b2     = (const float*)d_in[9];
  const float* g_w3     = (const float*)d_in[10];
  const float* g_b3     = (const float*)d_in[11];
  const float* g_w4     = (const float*)d_in[12];
  const float* g_b4     = (const float*)d_in[13];
  const float* g_skip_w = (const float*)d_in[14];
  const float* g_skip_b = (const float*)d_in[15];
  const float* ln_gamma = (const float*)d_in[16];
  const float* ln_beta  = (const float*)d_in[17];
  const float* wf = (const float*)d_in[18]; const float* bf = (const float*)d_in[19];
  const float* wi = (const float*)d_in[20]; const float* bi = (const float*)d_in[21];
  const float* wc = (const float*)d_in[22]; const float* bc = (const float*)d_in[23];
  const float* wo = (const float*)d_in[24]; const float* bo = (const float*)d_in[25];
  float* out = (float*)d_out;

  float* ws = (float*)d_ws;
  size_t o = 0;
  float* weighted = ws + o; o += (size_t)B_ * S_ * F_;
  float* P        = ws + o; o += (size_t)B_ * T_ * F_;
  float* D        = ws + o; o += (size_t)B_ * T_ * F_;
  float* sig      = ws + o; o += (size_t)B_ * SIGDIM;
  float* Baug     = ws + o; o += (size_t)KAUG * U_;
  float* partials = ws + o; o += (size_t)(B_ * 32) * U_;
  float* gemmPart = ws + o; o += (size_t)KSPLIT * B_ * U_;
  float* h1       = ws + o; o += (size_t)B_ * U_;
  float* ysig     = ws + o; o += (size_t)B_ * U_;
  float* h2       = ws + o; o += (size_t)B_ * U_;
  float* t3       = ws + o; o += (size_t)B_ * U_;
  float* t4       = ws + o; o += (size_t)B_ * U_;
  float* attn     = ws + o; o += (size_t)B_ * U_;
  float* current  = ws + o; o += (size_t)B_ * U_;
  float* comb     = ws + o; o += (size_t)B_ * 2 * U_;
  float* gates    = ws + o; o += (size_t)4 * B_ * U_;

  int n;
  n = B_ * S_ * F_;
  k_weighted<<<(n + 255) / 256, 256, 0, stream>>>(inp, tk, weighted, n);

  n = B_ * F_;
  k_pd<<<(n + 255) / 256, 256, 0, stream>>>(weighted, P, D, sig);

  k_sig2<<<dim3(F_ / 16, F_ / 16, B_), 32, 0, stream>>>(P, D, sig);

  n = KAUG * U_;
  k_baug<<<(n + 255) / 256, 256, 0, stream>>>(base_w, spline_w, Baug);

  // GRKAN dense layers (all f32 WMMA)
  dim3 gTile(U_ / 16, B_ / 16);
  dim3 gSplit(U_ / 16, B_ / 16, KSPLIT);
  n = B_ * U_;
  k_gemm_split<<<gSplit, 32, 0, stream>>>(sig, SIGDIM, g_w1, U_, gemmPart);
  k_combine<<<(n + 255) / 256, 256, 0, stream>>>(gemmPart, g_b1, h1, 1);
  k_gemm_split<<<gSplit, 32, 0, stream>>>(sig, SIGDIM, g_skip_w, U_, gemmPart);
  k_combine<<<(n + 255) / 256, 256, 0, stream>>>(gemmPart, g_skip_b, ysig, 0);
  k_gemm<<<gTile, 32, 0, stream>>>(h1, U_, g_w2, U_, g_b2, h2, U_, U_, 0);
  k_gemm<<<gTile, 32, 0, stream>>>(h2, U_, g_w3, U_, g_b3, t3, U_, U_, 2);
  k_gemm<<<gTile, 32, 0, stream>>>(h2, U_, g_w4, U_, g_b4, t4, U_, U_, 0);

  k_grkan_fin<<<B_, 256, 0, stream>>>(ysig, t3, t4, ln_gamma, ln_beta, attn);

  // fused KAN + attended-mean (dominant GEMM): 2048 row-tiles
  size_t lds = (size_t)(16 * KAUG + 16) * sizeof(float);
  k_kan<<<(B_ * S_) / 16, 256, lds, stream>>>(weighted, Baug, attn, partials);

  n = B_ * U_;
  k_reduce<<<(n + 255) / 256, 256, 0, stream>>>(partials, current);

  n = B_ * 2 * U_;
  k_combined<<<(n + 255) / 256, 256, 0, stream>>>(current, h_prev, comb);

  // LSTM gates: sigmoid(f,i,o), tanh(cand), fused into one launch
  k_gates<<<dim3(U_ / 16, B_ / 16, 4), 32, 0, stream>>>(
      comb, wf, wi, wc, wo, bf, bi, bc, bo, gates);

  n = B_ * U_;
  k_lstm<<<(n + 255) / 256, 256, 0, stream>>>(gates, c_prev, out);
}